// MessagePassing_52012053954612
// MI455X (gfx1250) — compile-verified
//
#include <hip/hip_runtime.h>

// MessagePassing fused kernel for MI455X (gfx1250, wave32).
//
// out[N][257] = concat(node_features[N][128],
//                      adj @ node_features   [N][128],
//                      diag_i = sum_j adj[i][j]*edge[j][i]   [N][1])
//
// Roofline: adj (1 GiB) + edge (1 GiB) streamed exactly once (NT-hint loads)
// at 23.3 TB/s -> ~92 us floor; node_features (8 MB) stays in the 192 MB L2.
// f32 WMMA 16x16x4 keeps full reference precision; kernel is memory-bound,
// so the inner loop is tuned for minimal instruction issue:
//   - k-paired LDS layout: each WMMA B operand is one contiguous b64 ->
//     compiler emits ds_load_2addr_b64 (2 operands/instruction, zero v_movs)
//   - pair-row stride padded to 288 dwords (== 32 mod 64 banks): lane halves
//     (t even/odd) map to disjoint bank halves -> conflict-free operand reads
//   - double-buffered staging, ONE barrier per 32-K chunk; next chunk's
//     global loads are issued before the barrier and overlap WMMA compute.

#define N_NODES 16384
#define F_DIM   128
#define OUT_W   257               // 2*F + 1
#define TK      32                // K-chunk staged in LDS
#define TM_WG   64                // rows per workgroup (4 waves x 16)
#define NWAVES  4
#define BLOCK   (NWAVES * 32)
#define NCHUNK  (N_NODES / TK)
#define PR_STRIDE 288             // dwords per k-pair row: 256 data + 32 pad

typedef __attribute__((ext_vector_type(2))) float v2f;
typedef __attribute__((ext_vector_type(4))) float v4f;
typedef __attribute__((ext_vector_type(8))) float v8f;

__global__ __launch_bounds__(BLOCK)
void mp_fused_kernel(const float* __restrict__ node,   // [N,128]
                     const float* __restrict__ edge,   // [N,N]
                     const float* __restrict__ adj,    // [N,N]
                     float* __restrict__ out)          // [N,257]
{
    // Paired layout: ldsB[buf][t*288 + c*2 + p] = node[k0 + 2t + p][c].
    // 2 x 18 KB double buffer (36 KB of 320 KB/WGP).
    __shared__ float ldsB[2][(TK / 2) * PR_STRIDE];

    const int tid    = threadIdx.x;
    const int lane   = tid & 31;
    const int wave   = tid >> 5;
    const int m      = lane & 15;      // row in M-tile / col in N-tile
    const int half   = lane >> 4;      // K-pair selector per ISA A/B layout
    const int wgBase = blockIdx.x * TM_WG;
    const int rowW   = wgBase + wave * 16;
    const int row    = rowW + m;

    // ---- pass-through copy: out[:, 0:128] = node_features (L2-resident)
    for (int i = tid; i < TM_WG * (F_DIM / 4); i += BLOCK) {
        const int r = wgBase + (i >> 5);
        const int c = (i & 31) * 4;
        v4f nv = *(const v4f*)(node + (size_t)r * F_DIM + c);
        float* o = out + (size_t)r * OUT_W + c;   // row stride 257 -> scalar stores
        o[0] = nv.x; o[1] = nv.y; o[2] = nv.z; o[3] = nv.w;
    }

    const float* adjRow = adj + (size_t)row * N_NODES;

    v8f acc[8] = {};        // 16x128 f32 C tile = 8 accumulators (64 VGPRs)
    float diag = 0.0f;

    // ---- staging pipeline registers: one 32x128 chunk = 8 float4 per thread
    v4f stg[8];
    auto stage_load = [&](int k0) {
        #pragma unroll
        for (int it = 0; it < 8; ++it) {
            const int slot = tid + it * BLOCK;     // float4 slot
            const int k    = slot >> 5;            // 32 float4 per k-row
            const int c4   = (slot & 31) * 4;
            stg[it] = *(const v4f*)(node + (size_t)(k0 + k) * F_DIM + c4);
        }
    };
    auto stage_store = [&](int bsel) {
        float* b = &ldsB[bsel][0];
        #pragma unroll
        for (int it = 0; it < 8; ++it) {
            const int slot = tid + it * BLOCK;
            const int k    = slot >> 5;
            const int c4   = (slot & 31) * 4;
            float* dst = b + (k >> 1) * PR_STRIDE + c4 * 2 + (k & 1);
            dst[0] = stg[it].x; dst[2] = stg[it].y;
            dst[4] = stg[it].z; dst[6] = stg[it].w;
        }
    };

    // prologue: chunk 0 into buffer 0
    stage_load(0);
    stage_store(0);

    for (int ci = 0; ci < NCHUNK; ++ci) {
        const int k0 = ci * TK;

        // Issue next chunk's global loads early: they overlap barrier+compute.
        if (ci + 1 < NCHUNK) stage_load(k0 + TK);

        // Single barrier per chunk: publishes buffer ci&1 (stores of iter
        // ci-1) AND fences reads of buffer (ci+1)&1 from compute of ci-1.
        __syncthreads();

        const float* buf = ldsB[ci & 1];

        #pragma unroll
        for (int s = 0; s < TK / 4; ++s) {
            const int t = s * 2 + half;        // K-pair index (K = 2t, 2t+1)

            // A fragment: adj[row][k0+2t : +2] -- one global b64, streamed (NT)
            v2f a = __builtin_nontemporal_load((const v2f*)(adjRow + k0 + 2 * t));

            // Fused diagonal: same adj values x edge[k][row] (streamed once, NT)
            const float* ecol = edge + (size_t)(k0 + 2 * t) * N_NODES + row;
            const float e0 = __builtin_nontemporal_load(ecol);
            const float e1 = __builtin_nontemporal_load(ecol + N_NODES);
            diag = __builtin_fmaf(a.x, e0, diag);
            diag = __builtin_fmaf(a.y, e1, diag);

            // 8 N-tiles: each B operand is one aligned b64 in LDS; compiler
            // pairs them into ds_load_2addr_b64 (offsets nt*16 b64-units).
            const float* bbase = buf + t * PR_STRIDE + m * 2;
            #pragma unroll
            for (int nt = 0; nt < 8; ++nt) {
                v2f b = *(const v2f*)(bbase + nt * 32);
                acc[nt] = __builtin_amdgcn_wmma_f32_16x16x4_f32(
                    /*neg_a=*/false, a, /*neg_b=*/false, b,
                    /*c_mod=*/(short)0, acc[nt],
                    /*reuse_a=*/false, /*reuse_b=*/false);
            }
        }

        // Store next chunk into the other buffer (safe: every wave passed
        // this iteration's barrier after its last read of that buffer).
        if (ci + 1 < NCHUNK) stage_store((ci + 1) & 1);
    }

    // ---- store C per 16x16 f32 C/D layout: VGPR v -> row M=v (lanes 0-15) / v+8
    #pragma unroll
    for (int nt = 0; nt < 8; ++nt) {
        #pragma unroll
        for (int v = 0; v < 8; ++v) {
            const int r = rowW + v + half * 8;
            out[(size_t)r * OUT_W + F_DIM + nt * 16 + m] = acc[nt][v];
        }
    }

    // ---- diag: lane r and lane r+16 hold complementary K halves for row r
    diag += __shfl_xor(diag, 16, 32);
    if (half == 0) {
        out[(size_t)row * OUT_W + 2 * F_DIM] = diag;
    }
}

extern "C" void kernel_launch(void* const* d_in, const int* in_sizes, int n_in,
                              void* d_out, int out_size, void* d_ws, size_t ws_size,
                              hipStream_t stream) {
    const float* node = (const float*)d_in[0];   // [N,128]
    const float* edge = (const float*)d_in[1];   // [N,N]
    const float* adj  = (const float*)d_in[2];   // [N,N]
    float* out = (float*)d_out;                  // [N,257]

    dim3 grid(N_NODES / TM_WG);   // 256 workgroups
    dim3 block(BLOCK);            // 128 threads = 4 wave32
    mp_fused_kernel<<<grid, block, 0, stream>>>(node, edge, adj, out);
}